// QLSTMClassifier_65481071400949
// MI455X (gfx1250) — compile-verified
//
#include <hip/hip_runtime.h>

// ---------------------------------------------------------------------------
// QLSTM classifier, restructured for MI455X (gfx1250, wave32, WMMA):
//   z_t = x_t @ Wx + hx @ Wh + bg   with x 1024-dim, hx 10-dim
// => hoist the x-part out of the scan as one big WMMA GEMM (memory-bound,
//    268MB gather @ 23.3TB/s ~ 11.5us, so fp32 WMMA keeps full precision at
//    zero cost), leaving only a tiny shuffle-parallel recurrence.
// ---------------------------------------------------------------------------

typedef float v2f __attribute__((ext_vector_type(2)));
typedef float v8f __attribute__((ext_vector_type(8)));

#define S_DIM 256          // batch == seq_len
#define E_DIM 1024         // embedding dim
#define H_DIM 10           // hidden dim / n_qubits
#define T_DIM 50           // tagset size
#define NPAD  64           // 4 gates, each padded to a 16-wide WMMA N tile
#define WG_G  (1034 * 10)  // per-gate stride inside Wg (4, 1034, 10)

// ------------------------- fast device math (CDNA5 HW trans ops) ------------
__device__ __forceinline__ float hw_cos(float x) {
  // v_cos_f32 takes revolutions: multiply radians by 1/(2*pi)
  return __builtin_amdgcn_cosf(x * 0.15915494309189535f);
}
__device__ __forceinline__ float hw_sigmoid(float x) {
  float e = __builtin_amdgcn_exp2f(-1.4426950408889634f * x);
  return __builtin_amdgcn_rcpf(1.0f + e);
}
__device__ __forceinline__ float hw_tanh(float x) {
  // tanh(x) = 2*sigmoid(2x) - 1
  float e = __builtin_amdgcn_exp2f(-2.8853900817779268f * x);
  return fmaf(2.0f, __builtin_amdgcn_rcpf(1.0f + e), -1.0f);
}

// ---------------------------------------------------------------------------
// Kernel A: Zx[row][g*16+h] = emb_table[sentence[row]] . Wg[g][:1024][h]
// row = t*256 + r (flat over all scan steps).  One wave per 16-row M tile,
// 4 fp32 WMMA accumulators (one per gate tile), K streamed in steps of 4.
// A fragment: lane l holds rows m = l&15, K pair (k + 2*(l>>4)) + {0,1}.
// B fragment mirrors with n = l&15.  Pad cols h=10..15 read neighbouring
// Wg values: finite, multiplied into Zx pad columns that are never consumed.
// ---------------------------------------------------------------------------
__global__ __launch_bounds__(256, 1)
void qlstm_zx_gemm(const int* __restrict__ sent, const float* __restrict__ emb,
                   const float* __restrict__ Wg, float* __restrict__ Zx) {
  const int tid  = threadIdx.x;
  const int l    = tid & 31;
  const int tile = blockIdx.x * 8 + (tid >> 5);   // 512 blocks * 8 waves
  const int m    = l & 15;                        // row in tile / col in N tile
  const int kh   = l >> 4;                        // 0/1 -> K offset 2*kh
  const int row  = tile * 16 + m;
  const int idx  = sent[row];                     // embedding gather index

  const float* arow = emb + (size_t)idx * E_DIM + 2 * kh;
  // B(k, g*16+m) = Wg[g*WG_G + k*10 + m]
  const float* b0 = Wg + (size_t)(2 * kh) * H_DIM + m;

  v8f acc0 = {0.f,0.f,0.f,0.f,0.f,0.f,0.f,0.f};
  v8f acc1 = acc0, acc2 = acc0, acc3 = acc0;

#pragma unroll 4
  for (int k = 0; k < E_DIM; k += 4) {
    const v2f a = *(const v2f*)(arow + k);        // global_load_b64 (L2/HBM)
    const float* bk = b0 + (size_t)k * H_DIM;     // L2-resident weights
    v2f bf0, bf1, bf2, bf3;
    bf0.x = bk[0];             bf0.y = bk[H_DIM];
    bf1.x = bk[WG_G];          bf1.y = bk[WG_G + H_DIM];
    bf2.x = bk[2 * WG_G];      bf2.y = bk[2 * WG_G + H_DIM];
    bf3.x = bk[3 * WG_G];      bf3.y = bk[3 * WG_G + H_DIM];
    acc0 = __builtin_amdgcn_wmma_f32_16x16x4_f32(false, a, false, bf0,
                                                 (short)0, acc0, false, false);
    acc1 = __builtin_amdgcn_wmma_f32_16x16x4_f32(false, a, false, bf1,
                                                 (short)0, acc1, false, false);
    acc2 = __builtin_amdgcn_wmma_f32_16x16x4_f32(false, a, false, bf2,
                                                 (short)0, acc2, false, false);
    acc3 = __builtin_amdgcn_wmma_f32_16x16x4_f32(false, a, false, bf3,
                                                 (short)0, acc3, false, false);
  }

  // C layout: VGPR v, lanes 0-15 -> M=v, lanes 16-31 -> M=v+8, N = l&15
  float* zb = Zx + (size_t)(tile * 16 + 8 * kh) * NPAD + m;
#pragma unroll
  for (int v = 0; v < 8; ++v) {
    zb[(size_t)v * NPAD +  0] = acc0[v];
    zb[(size_t)v * NPAD + 16] = acc1[v];
    zb[(size_t)v * NPAD + 32] = acc2[v];
    zb[(size_t)v * NPAD + 48] = acc3[v];
  }
}

// ---------------------------------------------------------------------------
// Kernel B: the sequential scan.  Lane = (row, wire h), 16 lanes per row
// (h=10..15 are finite dead lanes), 2 rows per wave32.  hx/cx and the whole
// 10x4 Wh slice live in registers; hx broadcasts and the per-gate prefix
// products (cumprod) are wave shuffles (ds_bpermute), no LDS round trips.
// qlayer: out_0 = prod_{j>=1} c_j ; out_w = prod_{j<=w} c_j, computed from a
// single inclusive scan with s_0 := 1 (no division by cos).
// ---------------------------------------------------------------------------
__global__ __launch_bounds__(256, 1)
void qlstm_recurrence(const float* __restrict__ Zx, const float* __restrict__ Wg,
                      const float* __restrict__ bg, const float* __restrict__ theta,
                      float* __restrict__ outs) {
  const int tid = threadIdx.x;
  const int h   = tid & 15;
  const int row = blockIdx.x * 16 + (tid >> 4);
  const int l   = tid & 31;
  const int rb  = l & 16;                         // row-group base lane
  const int hc  = (h < H_DIM) ? h : (H_DIM - 1);  // clamp for dead lanes

  float wh[10][4];                                // Wh[j][g] for this wire
  float bt[4];                                    // bg + theta folded
#pragma unroll
  for (int g = 0; g < 4; ++g) {
    bt[g] = bg[g * H_DIM + hc] + theta[g * H_DIM + hc];
#pragma unroll
    for (int j = 0; j < H_DIM; ++j)
      wh[j][g] = Wg[(size_t)g * WG_G + (size_t)(E_DIM + j) * H_DIM + hc];
  }

  float hx = 0.0f, cx = 0.0f;
  for (int t = 0; t < S_DIM; ++t) {
    const float* zr = Zx + ((size_t)t * S_DIM + row) * NPAD;
    float z0 = zr[h]      + bt[0];
    float z1 = zr[16 + h] + bt[1];
    float z2 = zr[32 + h] + bt[2];
    float z3 = zr[48 + h] + bt[3];
#pragma unroll
    for (int j = 0; j < H_DIM; ++j) {             // hx @ Wh (10x40)
      float hj = __shfl(hx, rb + j, 32);
      z0 = fmaf(hj, wh[j][0], z0);
      z1 = fmaf(hj, wh[j][1], z1);
      z2 = fmaf(hj, wh[j][2], z2);
      z3 = fmaf(hj, wh[j][3], z3);
    }
    float carr[4] = { hw_cos(z0), hw_cos(z1), hw_cos(z2), hw_cos(z3) };
    float outg[4];
#pragma unroll
    for (int g = 0; g < 4; ++g) {
      float Q = (h == 0) ? 1.0f : carr[g];        // s_0 = 1 trick
#pragma unroll
      for (int hop = 1; hop <= 8; hop <<= 1) {    // inclusive prefix product
        float v = __shfl(Q, rb + ((h >= hop) ? (h - hop) : 0), 32);
        Q = (h >= hop) ? Q * v : Q;
      }
      float q9  = __shfl(Q, rb + 9, 32);          // prod_{1..9} c_j
      float cc0 = __shfl(carr[g], rb, 32);        // c_0
      outg[g] = (h == 0) ? q9 : cc0 * Q;
    }
    float f  = hw_sigmoid(outg[0]);
    float i  = hw_sigmoid(outg[1]);
    float gg = hw_tanh(outg[2]);
    float o  = hw_sigmoid(outg[3]);
    cx = fmaf(f, cx, i * gg);
    hx = o * hw_tanh(cx);
    if (h < H_DIM)
      outs[((size_t)t * S_DIM + row) * H_DIM + h] = hx;
  }
}

// ---------------------------------------------------------------------------
// Kernel C: logits = outs @ W_out + b_out, then log_softmax over T=50.
// One thread per (t,row); W_out/b_out broadcast from LDS.
// ---------------------------------------------------------------------------
__global__ __launch_bounds__(256, 1)
void qlstm_logits(const float* __restrict__ outs, const float* __restrict__ Wout,
                  const float* __restrict__ bout, float* __restrict__ out) {
  __shared__ float Ws[H_DIM * T_DIM];
  __shared__ float Bs[T_DIM];
  for (int i = threadIdx.x; i < H_DIM * T_DIM; i += 256) Ws[i] = Wout[i];
  for (int i = threadIdx.x; i < T_DIM; i += 256) Bs[i] = bout[i];
  __syncthreads();

  const int row = blockIdx.x * 256 + threadIdx.x;
  float hv[H_DIM];
#pragma unroll
  for (int j = 0; j < H_DIM; ++j) hv[j] = outs[(size_t)row * H_DIM + j];

  float lg[T_DIM];
  float mx = -3.4e38f;
#pragma unroll
  for (int c = 0; c < T_DIM; ++c) {
    float a = Bs[c];
#pragma unroll
    for (int j = 0; j < H_DIM; ++j) a = fmaf(hv[j], Ws[j * T_DIM + c], a);
    lg[c] = a;
    mx = fmaxf(mx, a);
  }
  float sum = 0.0f;
#pragma unroll
  for (int c = 0; c < T_DIM; ++c)
    sum += __builtin_amdgcn_exp2f((lg[c] - mx) * 1.4426950408889634f);
  float lns = __builtin_amdgcn_logf(sum) * 0.69314718055994531f;

  float* orow = out + (size_t)row * T_DIM;
#pragma unroll
  for (int c = 0; c < T_DIM; ++c) orow[c] = (lg[c] - mx) - lns;
}

// ---------------------------------------------------------------------------
extern "C" void kernel_launch(void* const* d_in, const int* in_sizes, int n_in,
                              void* d_out, int out_size, void* d_ws, size_t ws_size,
                              hipStream_t stream) {
  (void)in_sizes; (void)n_in; (void)out_size; (void)ws_size;
  const int*   sent  = (const int*)  d_in[0];   // (256,256) int32
  const float* emb   = (const float*)d_in[1];   // (50257,1024)
  const float* Wg    = (const float*)d_in[2];   // (4,1034,10)
  const float* bg    = (const float*)d_in[3];   // (4,10)
  const float* theta = (const float*)d_in[4];   // (4,10)
  const float* Wout  = (const float*)d_in[5];   // (10,50)
  const float* bout  = (const float*)d_in[6];   // (50,)
  float* outp = (float*)d_out;                  // (256,256,50) f32

  float* Zx   = (float*)d_ws;                   // 65536 x 64 f32 (16.8 MB)
  float* outs = Zx + (size_t)65536 * NPAD;      // 65536 x 10 f32 ( 2.6 MB)

  qlstm_zx_gemm   <<<512, 256, 0, stream>>>(sent, emb, Wg, Zx);
  qlstm_recurrence<<< 16, 256, 0, stream>>>(Zx, Wg, bg, theta, outs);
  qlstm_logits    <<<256, 256, 0, stream>>>(outs, Wout, bout, outp);
}